// GraphConditioner_20615843021322
// MI455X (gfx1250) — compile-verified
//
#include <hip/hip_runtime.h>
#include <hip/hip_bf16.h>
#include <stdint.h>

typedef float v2f __attribute__((ext_vector_type(2)));
typedef float v8f __attribute__((ext_vector_type(8)));

#define WB 8     // waves per 256-thread block (wave32)
#define SLOPE 0.2f
#define EPS 1e-5f

__device__ __forceinline__ float waveSum(float v) {
#pragma unroll
  for (int off = 16; off > 0; off >>= 1) v += __shfl_xor(v, off, 32);
  return v;
}

// monotone encoding of float for unsigned atomicMax
__device__ __forceinline__ unsigned fenc(float f) {
  unsigned u = __float_as_uint(f);
  return (u & 0x80000000u) ? ~u : (u | 0x80000000u);
}
__device__ __forceinline__ float fdec(unsigned u) {
  u = (u & 0x80000000u) ? (u & 0x7fffffffu) : ~u;
  return __uint_as_float(u);
}

// ---------------------------------------------------------------- utilities
__global__ void k_fill_u32(unsigned* __restrict__ p, unsigned v, int n) {
  int i = blockIdx.x * blockDim.x + threadIdx.x;
  if (i < n) p[i] = v;
}

__global__ void k_count(const int* __restrict__ batch, float* __restrict__ gcnt, int N) {
  int i = blockIdx.x * blockDim.x + threadIdx.x;
  if (i < N) atomicAdd(gcnt + batch[i], 1.0f);
}

// --------------------------------------------- token attention pooling
// one block (256 thr) per node; stage the 16x768 tile in LDS, read hidden ONCE
__global__ void k_token_pool(const float* __restrict__ hidden,
                             const float* __restrict__ attn_w,
                             const float* __restrict__ attn_b,
                             float* __restrict__ x0) {
  __shared__ float lds[16 * 768];
  __shared__ float part[256];
  __shared__ float sc[16];
  const int n = blockIdx.x;
  const int t = threadIdx.x;
  const float* src = hidden + (size_t)n * (16 * 768);
#pragma unroll
  for (int k = 0; k < 48; ++k) lds[t + k * 256] = src[t + k * 256];
  __syncthreads();
  // scores[r] = dot(hidden[n,r,:], attn_w) + b ; 16 threads per row
  const int r = t >> 4, j = t & 15;
  float p = 0.f;
  const int h0 = j * 48;
#pragma unroll 8
  for (int h = 0; h < 48; ++h) p += lds[r * 768 + h0 + h] * attn_w[h0 + h];
  part[t] = p;
  __syncthreads();
  if (t < 16) {
    float s = attn_b[0];
#pragma unroll
    for (int q = 0; q < 16; ++q) s += part[t * 16 + q];
    sc[t] = s;
  }
  __syncthreads();
  // x0[n,h] = sum_l hidden[n,l,h] * score[l]
  float* dst = x0 + (size_t)n * 768;
#pragma unroll
  for (int m = 0; m < 3; ++m) {
    const int hh = t + m * 256;
    float acc = 0.f;
#pragma unroll
    for (int l = 0; l < 16; ++l) acc += lds[l * 768 + hh] * sc[l];
    dst[hh] = acc;
  }
}

// --------------------------------------------- WMMA f32 GEMM: C = A * W^T
// A:[M,K] row-major, W:[Nc,K] row-major, C:[M,Nc]. One wave per 16x16 tile.
__global__ void k_gemm_wmma(const float* __restrict__ A,
                            const float* __restrict__ W,
                            float* __restrict__ C,
                            int M, int K, int Nc) {
  const int tid  = blockIdx.x * WB + (threadIdx.x >> 5);
  const int lane = threadIdx.x & 31;
  const int tilesN = Nc >> 4;
  const int tilesM = M >> 4;
  if (tid >= tilesM * tilesN) return;          // wave-uniform exit
  const int mt = tid / tilesN, nt = tid % tilesN;
  const int row  = (mt << 4) + (lane & 15);    // A: lanes 0-15 & 16-31 -> M=0..15
  const int col  = (nt << 4) + (lane & 15);    // B: lanes -> N=0..15
  const int koff = (lane >> 4) << 1;           // upper half-wave carries K+2,K+3
  const float* ap = A + (size_t)row * K + koff;
  const float* bp = W + (size_t)col * K + koff;
  v8f acc = {};
#pragma unroll 4
  for (int k = 0; k < K; k += 4) {
    v2f a = *(const v2f*)(ap + k);             // {A[row][k+koff], A[row][k+koff+1]}
    v2f b = *(const v2f*)(bp + k);             // {W[col][k+koff], W[col][k+koff+1]}
    acc = __builtin_amdgcn_wmma_f32_16x16x4_f32(
        /*neg_a=*/false, a, /*neg_b=*/false, b,
        /*c_mod=*/(short)0, acc, /*reuse_a=*/false, /*reuse_b=*/false);
  }
  // C/D layout: VGPR j -> row m0+j (lanes 0-15) / m0+8+j (lanes 16-31)
  const int cr = (mt << 4) + ((lane >> 4) << 3);
  const int cc = (nt << 4) + (lane & 15);
  float* cp = C + (size_t)cr * Nc + cc;
#pragma unroll
  for (int jj = 0; jj < 8; ++jj) cp[(size_t)jj * Nc] = acc[jj];
}

// --------------------------------------------- per-node dual dot products
__global__ void k_dot2(const float* __restrict__ h, const float* __restrict__ a1,
                       const float* __restrict__ a2, float* __restrict__ o1,
                       float* __restrict__ o2, int N, int D) {
  const int w = blockIdx.x * WB + (threadIdx.x >> 5);
  const int lane = threadIdx.x & 31;
  if (w >= N) return;
  const float* hp = h + (size_t)w * D;
  float s1 = 0.f, s2 = 0.f;
  for (int c = lane; c < D; c += 32) {
    float v = hp[c];
    s1 += v * a1[c];
    s2 += v * a2[c];
  }
  s1 = waveSum(s1);
  s2 = waveSum(s2);
  if (lane == 0) { o1[w] = s1; o2[w] = s2; }
}

__global__ void k_dot1(const float* __restrict__ x, const float* __restrict__ w,
                       const float* __restrict__ b, float* __restrict__ o, int N, int D) {
  const int n = blockIdx.x * WB + (threadIdx.x >> 5);
  const int lane = threadIdx.x & 31;
  if (n >= N) return;
  const float* xp = x + (size_t)n * D;
  float s = 0.f;
  for (int c = lane; c < D; c += 32) s += xp[c] * w[c];
  s = waveSum(s);
  if (lane == 0) o[n] = s + b[0];
}

// --------------------------------------------- edge softmax (3 passes)
__global__ void k_edge_max(const float* __restrict__ hs, const float* __restrict__ hd,
                           const int* __restrict__ src, const int* __restrict__ dst,
                           int E, int N, unsigned* __restrict__ encmax) {
  const int i = blockIdx.x * blockDim.x + threadIdx.x;
  if (i >= E + N) return;
  int s, d;
  if (i < E) { s = src[i]; d = dst[i]; } else { s = d = i - E; }
  float e = hs[s] + hd[d];
  e = e > 0.f ? e : SLOPE * e;
  atomicMax(encmax + d, fenc(e));
}

__global__ void k_edge_expsum(const float* __restrict__ hs, const float* __restrict__ hd,
                              const int* __restrict__ src, const int* __restrict__ dst,
                              int E, int N, const unsigned* __restrict__ encmax,
                              float* __restrict__ ew, float* __restrict__ ssum) {
  const int i = blockIdx.x * blockDim.x + threadIdx.x;
  if (i >= E + N) return;
  int s, d;
  if (i < E) { s = src[i]; d = dst[i]; } else { s = d = i - E; }
  float e = hs[s] + hd[d];
  e = e > 0.f ? e : SLOPE * e;
  float w = __expf(e - fdec(encmax[d]));
  ew[i] = w;
  atomicAdd(ssum + d, w);
}

__global__ void k_init_bias(const float* __restrict__ b, float* __restrict__ out, int total, int D) {
  const int i = blockIdx.x * blockDim.x + threadIdx.x;
  if (i < total) out[i] = b[i & (D - 1)];
}

// one wave per edge: out[dst] += alpha * h[src]
__global__ void k_edge_aggr(const float* __restrict__ h, const float* __restrict__ ew,
                            const float* __restrict__ ssum, const int* __restrict__ src,
                            const int* __restrict__ dst, int E, int N,
                            float* __restrict__ out, int D) {
  const int i = blockIdx.x * WB + (threadIdx.x >> 5);
  const int lane = threadIdx.x & 31;
  if (i >= E + N) return;
  int s, d;
  if (i < E) { s = src[i]; d = dst[i]; } else { s = d = i - E; }
  const float alpha = ew[i] / ssum[d];
  const float* hp = h + (size_t)s * D;
  float* op = out + (size_t)d * D;
  for (int c = lane; c < D; c += 32) atomicAdd(op + c, alpha * hp[c]);
}

// --------------------------------------------- graph layernorm (+ReLU)
__global__ void k_ln_sum(const float* __restrict__ x, const int* __restrict__ batch,
                         float* __restrict__ gsum, int N, int D) {
  const int n = blockIdx.x * WB + (threadIdx.x >> 5);
  const int lane = threadIdx.x & 31;
  if (n >= N) return;
  const float* xp = x + (size_t)n * D;
  float s = 0.f;
  for (int c = lane; c < D; c += 32) s += xp[c];
  s = waveSum(s);
  if (lane == 0) atomicAdd(gsum + batch[n], s);
}

__global__ void k_ln_mean(float* __restrict__ gsum, const float* __restrict__ gcnt, int G, int D) {
  const int g = blockIdx.x * blockDim.x + threadIdx.x;
  if (g < G) gsum[g] = gsum[g] / (fmaxf(gcnt[g], 1.f) * (float)D);
}

__global__ void k_ln_var(const float* __restrict__ x, const int* __restrict__ batch,
                         const float* __restrict__ gmean, float* __restrict__ gvar, int N, int D) {
  const int n = blockIdx.x * WB + (threadIdx.x >> 5);
  const int lane = threadIdx.x & 31;
  if (n >= N) return;
  const float m = gmean[batch[n]];
  const float* xp = x + (size_t)n * D;
  float s = 0.f;
  for (int c = lane; c < D; c += 32) {
    float v = xp[c] - m;
    s += v * v;
  }
  s = waveSum(s);
  if (lane == 0) atomicAdd(gvar + batch[n], s);
}

__global__ void k_ln_rstd(float* __restrict__ gvar, const float* __restrict__ gcnt, int G, int D) {
  const int g = blockIdx.x * blockDim.x + threadIdx.x;
  if (g < G) gvar[g] = rsqrtf(gvar[g] / (fmaxf(gcnt[g], 1.f) * (float)D) + EPS);
}

__global__ void k_ln_apply(const float* __restrict__ x, const int* __restrict__ batch,
                           const float* __restrict__ gmean, const float* __restrict__ grstd,
                           const float* __restrict__ w, const float* __restrict__ bb,
                           float* __restrict__ y, int N, int D) {
  const int i = blockIdx.x * blockDim.x + threadIdx.x;
  if (i >= N * D) return;
  const int n = i / D, c = i & (D - 1);
  const int g = batch[n];
  const float v = (x[i] - gmean[g]) * grstd[g] * w[c] + bb[c];
  y[i] = fmaxf(v, 0.f);
}

// --------------------------------------------- global attention pooling
__global__ void k_seg_max(const float* __restrict__ gl, const int* __restrict__ batch,
                          unsigned* __restrict__ enc, int N) {
  const int i = blockIdx.x * blockDim.x + threadIdx.x;
  if (i < N) atomicMax(enc + batch[i], fenc(gl[i]));
}

__global__ void k_seg_expsum(const float* __restrict__ gl, const int* __restrict__ batch,
                             const unsigned* __restrict__ enc, float* __restrict__ ge,
                             float* __restrict__ gs, int N) {
  const int i = blockIdx.x * blockDim.x + threadIdx.x;
  if (i >= N) return;
  const int g = batch[i];
  const float e = __expf(gl[i] - fdec(enc[g]));
  ge[i] = e;
  atomicAdd(gs + g, e);
}

__global__ void k_pool(const float* __restrict__ x, const float* __restrict__ ge,
                       const float* __restrict__ gs, const int* __restrict__ batch,
                       float* __restrict__ out, int N, int D) {
  const int n = blockIdx.x * WB + (threadIdx.x >> 5);
  const int lane = threadIdx.x & 31;
  if (n >= N) return;
  const int g = batch[n];
  const float coef = ge[n] / gs[g];
  const float* xp = x + (size_t)n * D;
  float* op = out + (size_t)g * D;
  for (int c = lane; c < D; c += 32) atomicAdd(op + c, coef * xp[c]);
}

// ----------------------------------------------------------------- driver
static inline int cdiv(int a, int b) { return (a + b - 1) / b; }

extern "C" void kernel_launch(void* const* d_in, const int* in_sizes, int n_in,
                              void* d_out, int out_size, void* d_ws, size_t ws_size,
                              hipStream_t stream) {
  const float* hidden = (const float*)d_in[0];
  const float* attn_w = (const float*)d_in[1];
  const float* attn_b = (const float*)d_in[2];
  const float* W1  = (const float*)d_in[3];
  const float* as1 = (const float*)d_in[4];
  const float* ad1 = (const float*)d_in[5];
  const float* b1  = (const float*)d_in[6];
  const float* lw1 = (const float*)d_in[7];
  const float* lb1 = (const float*)d_in[8];
  const float* W2  = (const float*)d_in[9];
  const float* as2 = (const float*)d_in[10];
  const float* ad2 = (const float*)d_in[11];
  const float* b2  = (const float*)d_in[12];
  const float* lw2 = (const float*)d_in[13];
  const float* lb2 = (const float*)d_in[14];
  const float* gate_w = (const float*)d_in[15];
  const float* gate_b = (const float*)d_in[16];
  const int* ei    = (const int*)d_in[17];
  const int* batch = (const int*)d_in[18];

  const int N = in_sizes[18];
  const int E = in_sizes[17] / 2;
  const int H = 768, D = 256;
  const int G = out_size / D;
  const int Et = E + N;
  const int* esrc = ei;
  const int* edst = ei + E;

  // workspace carve-up (floats)
  float* ws = (float*)d_ws;
  size_t o = 0;
  float*    xbig = ws + o; o += (size_t)N * H;   // x0, then reused for x1/x2
  float*    hbuf = ws + o; o += (size_t)N * D;   // GEMM output h (both layers)
  float*    agg  = ws + o; o += (size_t)N * D;   // GAT aggregation output
  float*    hs   = ws + o; o += (size_t)N;       // h @ a_s   (also gate logits)
  float*    hd   = ws + o; o += (size_t)N;       // h @ a_d   (also gate exp)
  unsigned* enc  = (unsigned*)(ws + o); o += (size_t)N;  // per-dst running max
  float*    ssum = ws + o; o += (size_t)N;       // per-dst exp-sum
  float*    ew   = ws + o; o += (size_t)Et;      // per-edge exp weight
  float*    gcnt = ws + o; o += (size_t)G;       // nodes per graph
  float*    gmv  = ws + o; o += (size_t)G;       // graph sum -> mean
  float*    gvr  = ws + o; o += (size_t)G;       // graph var -> rstd
  unsigned* genc = (unsigned*)(ws + o); o += (size_t)G;  // pooling max
  float*    gse  = ws + o; o += (size_t)G;       // pooling exp-sum
  (void)ws_size; (void)n_in;

  const int nodeWaveBlocks = cdiv(N, WB);
  const int edgeWaveBlocks = cdiv(Et, WB);
  const int tiles = (N / 16) * (D / 16);

  // graph node counts (used by both LN layers)
  k_fill_u32<<<cdiv(G, 256), 256, 0, stream>>>((unsigned*)gcnt, 0u, G);
  k_count<<<cdiv(N, 256), 256, 0, stream>>>(batch, gcnt, N);

  // 1) token attention pooling: hidden -> x0  (single HBM pass over hidden)
  k_token_pool<<<N, 256, 0, stream>>>(hidden, attn_w, attn_b, xbig);

  // ================= GAT layer 1 =================
  k_gemm_wmma<<<cdiv(tiles, WB), 256, 0, stream>>>(xbig, W1, hbuf, N, H, D);
  k_dot2<<<nodeWaveBlocks, 256, 0, stream>>>(hbuf, as1, ad1, hs, hd, N, D);
  k_fill_u32<<<cdiv(N, 256), 256, 0, stream>>>(enc, 0u, N);
  k_fill_u32<<<cdiv(N, 256), 256, 0, stream>>>((unsigned*)ssum, 0u, N);
  k_edge_max<<<cdiv(Et, 256), 256, 0, stream>>>(hs, hd, esrc, edst, E, N, enc);
  k_edge_expsum<<<cdiv(Et, 256), 256, 0, stream>>>(hs, hd, esrc, edst, E, N, enc, ew, ssum);
  k_init_bias<<<cdiv(N * D, 256), 256, 0, stream>>>(b1, agg, N * D, D);
  k_edge_aggr<<<edgeWaveBlocks, 256, 0, stream>>>(hbuf, ew, ssum, esrc, edst, E, N, agg, D);
  // graph LN + ReLU -> x1 (reuse xbig)
  k_fill_u32<<<cdiv(G, 256), 256, 0, stream>>>((unsigned*)gmv, 0u, G);
  k_fill_u32<<<cdiv(G, 256), 256, 0, stream>>>((unsigned*)gvr, 0u, G);
  k_ln_sum<<<nodeWaveBlocks, 256, 0, stream>>>(agg, batch, gmv, N, D);
  k_ln_mean<<<cdiv(G, 256), 256, 0, stream>>>(gmv, gcnt, G, D);
  k_ln_var<<<nodeWaveBlocks, 256, 0, stream>>>(agg, batch, gmv, gvr, N, D);
  k_ln_rstd<<<cdiv(G, 256), 256, 0, stream>>>(gvr, gcnt, G, D);
  k_ln_apply<<<cdiv(N * D, 256), 256, 0, stream>>>(agg, batch, gmv, gvr, lw1, lb1, xbig, N, D);

  // ================= GAT layer 2 =================
  k_gemm_wmma<<<cdiv(tiles, WB), 256, 0, stream>>>(xbig, W2, hbuf, N, D, D);
  k_dot2<<<nodeWaveBlocks, 256, 0, stream>>>(hbuf, as2, ad2, hs, hd, N, D);
  k_fill_u32<<<cdiv(N, 256), 256, 0, stream>>>(enc, 0u, N);
  k_fill_u32<<<cdiv(N, 256), 256, 0, stream>>>((unsigned*)ssum, 0u, N);
  k_edge_max<<<cdiv(Et, 256), 256, 0, stream>>>(hs, hd, esrc, edst, E, N, enc);
  k_edge_expsum<<<cdiv(Et, 256), 256, 0, stream>>>(hs, hd, esrc, edst, E, N, enc, ew, ssum);
  k_init_bias<<<cdiv(N * D, 256), 256, 0, stream>>>(b2, agg, N * D, D);
  k_edge_aggr<<<edgeWaveBlocks, 256, 0, stream>>>(hbuf, ew, ssum, esrc, edst, E, N, agg, D);
  // graph LN + ReLU -> x2 (reuse xbig)
  k_fill_u32<<<cdiv(G, 256), 256, 0, stream>>>((unsigned*)gmv, 0u, G);
  k_fill_u32<<<cdiv(G, 256), 256, 0, stream>>>((unsigned*)gvr, 0u, G);
  k_ln_sum<<<nodeWaveBlocks, 256, 0, stream>>>(agg, batch, gmv, N, D);
  k_ln_mean<<<cdiv(G, 256), 256, 0, stream>>>(gmv, gcnt, G, D);
  k_ln_var<<<nodeWaveBlocks, 256, 0, stream>>>(agg, batch, gmv, gvr, N, D);
  k_ln_rstd<<<cdiv(G, 256), 256, 0, stream>>>(gvr, gcnt, G, D);
  k_ln_apply<<<cdiv(N * D, 256), 256, 0, stream>>>(agg, batch, gmv, gvr, lw2, lb2, xbig, N, D);

  // ================= global attention pooling =================
  k_dot1<<<nodeWaveBlocks, 256, 0, stream>>>(xbig, gate_w, gate_b, hs, N, D);
  k_fill_u32<<<cdiv(G, 256), 256, 0, stream>>>(genc, 0u, G);
  k_fill_u32<<<cdiv(G, 256), 256, 0, stream>>>((unsigned*)gse, 0u, G);
  k_seg_max<<<cdiv(N, 256), 256, 0, stream>>>(hs, batch, genc, N);
  k_seg_expsum<<<cdiv(N, 256), 256, 0, stream>>>(hs, batch, genc, hd, gse, N);
  k_fill_u32<<<cdiv(G * D, 256), 256, 0, stream>>>((unsigned*)d_out, 0u, G * D);
  k_pool<<<nodeWaveBlocks, 256, 0, stream>>>(xbig, hd, gse, batch, (float*)d_out, N, D);
}